// GIN_LCG_84370337563246
// MI455X (gfx1250) — compile-verified
//
#include <hip/hip_runtime.h>
#include <hip/hip_bf16.h>

// ---------------------------------------------------------------------------
// CDNA5 (gfx1250) WMMA types
// ---------------------------------------------------------------------------
typedef __attribute__((ext_vector_type(2))) float v2f;
typedef __attribute__((ext_vector_type(8))) float v8f;

#define D_DIM 256

// ---------------------------------------------------------------------------
// Tiled fp32 WMMA GEMM:  Y[M,256] = act(A[M,K] @ W[K,256] + bias)
//   - block tile 128x64, BK=16, 256 threads = 8 waves (4 in M x 2 in N)
//   - wave tile 32x32 -> 4 accumulators (v8f each)
//   - LDS tiles stored as float2 (K,K+1) pairs so every WMMA fragment is one
//     aligned ds_load_b64 (fp32 fragment layout: K = vgpr + 2*lane_half)
//   - double-buffered LDS: next chunk's global loads issue before compute,
//     ds_store + single barrier after compute (global latency hidden by WMMA)
//   - SWAP: A logical row r reads physical row r^1 (literal pair swap)
// ---------------------------------------------------------------------------
template <bool RELU, bool SWAP>
__global__ __launch_bounds__(256) void gemm_mlp(
    const float* __restrict__ A, const float* __restrict__ W,
    const float* __restrict__ bias, float* __restrict__ Y, int M, int K) {
  __shared__ float2 sA2[2][128][10];  // 2 x (128 rows x 8 k-pairs, +2 pad)
  __shared__ float2 sB2[2][8][66];    // 2 x (8 k-pairs x 64 cols, +2 pad)

  const int tid  = threadIdx.x;
  const int lane = tid & 31;
  const int wave = tid >> 5;
  const int wm   = wave & 3;   // 32-row slab
  const int wn   = wave >> 2;  // 32-col slab
  const int mblk = blockIdx.x * 128;
  const int nblk = blockIdx.y * 64;

  const int half = lane >> 4;  // ISA lane-half: selects K pair / M group
  const int l16  = lane & 15;

  v8f acc00 = {}, acc01 = {}, acc10 = {}, acc11 = {};

  // B staging map: thread -> k-pair (tid>>5), two cols (tid&31)*2
  const int bkp = tid >> 5;
  const int bn2 = (tid & 31) * 2;

  float4 ra[2];   // staged A (2 float4 per thread)
  float2 rb[2];   // staged B (rows k, k+1)

  auto load_tiles = [&](int k0) {
#pragma unroll
    for (int i = 0; i < 2; ++i) {
      const int idx = tid + i * 256;   // 0..511
      const int ar  = idx >> 2;        // 0..127
      const int ac4 = (idx & 3) << 2;  // 0,4,8,12
      const int row = mblk + ar;
      float4 v = make_float4(0.f, 0.f, 0.f, 0.f);
      if (row < M) {
        const int prow = SWAP ? (row ^ 1) : row;
        v = *(const float4*)(A + (size_t)prow * K + k0 + ac4);
      }
      ra[i] = v;
    }
    const float* p0 = W + (size_t)(k0 + bkp * 2) * D_DIM + nblk + bn2;
    rb[0] = *(const float2*)p0;           // row k
    rb[1] = *(const float2*)(p0 + D_DIM); // row k+1
  };

  auto store_tiles = [&](int buf) {
#pragma unroll
    for (int i = 0; i < 2; ++i) {
      const int idx = tid + i * 256;
      const int ar  = idx >> 2;
      const int ac4 = (idx & 3) << 2;
      *(float4*)&sA2[buf][ar][ac4 >> 1] = ra[i];  // 16B aligned (80B row)
    }
    float4 wv;
    wv.x = rb[0].x; wv.y = rb[1].x;  // (B[k][n],   B[k+1][n])
    wv.z = rb[0].y; wv.w = rb[1].y;  // (B[k][n+1], B[k+1][n+1])
    *(float4*)&sB2[buf][bkp][bn2] = wv;
  };

  // prologue: stage chunk 0
  load_tiles(0);
  store_tiles(0);
  __syncthreads();

  const int nChunks = K >> 4;
  for (int c = 0; c < nChunks; ++c) {
    const int cur = c & 1;
    const bool hasNext = (c + 1) < nChunks;

    if (hasNext) {
      // prefetch chunk c+2's B rows, then issue chunk c+1's loads (no wait:
      // s_wait_loadcnt lands after the 16 WMMAs below)
      __builtin_prefetch(
          W + (size_t)(((c + 2) << 4) + bkp * 2) * D_DIM + nblk + bn2, 0, 3);
      load_tiles((c + 1) << 4);
    }

#pragma unroll
    for (int kk = 0; kk < 4; ++kk) {
      const int kp = kk * 2 + half;  // this lane-half's (K,K+1) pair index
      const v2f a0 = *(const v2f*)&sA2[cur][wm * 32 + l16][kp];
      const v2f a1 = *(const v2f*)&sA2[cur][wm * 32 + 16 + l16][kp];
      const v2f b0 = *(const v2f*)&sB2[cur][kp][wn * 32 + l16];
      const v2f b1 = *(const v2f*)&sB2[cur][kp][wn * 32 + 16 + l16];
      acc00 = __builtin_amdgcn_wmma_f32_16x16x4_f32(false, a0, false, b0,
                                                    (short)0, acc00, false, false);
      acc01 = __builtin_amdgcn_wmma_f32_16x16x4_f32(false, a0, false, b1,
                                                    (short)0, acc01, false, false);
      acc10 = __builtin_amdgcn_wmma_f32_16x16x4_f32(false, a1, false, b0,
                                                    (short)0, acc10, false, false);
      acc11 = __builtin_amdgcn_wmma_f32_16x16x4_f32(false, a1, false, b1,
                                                    (short)0, acc11, false, false);
    }

    if (hasNext) {
      store_tiles(cur ^ 1);
      __syncthreads();  // uniform condition -> all waves hit same barriers
    }
  }

  // ---- epilogue: bias (+ReLU); C layout M = v + 8*half, N = lane%16 ----
  const int n0 = nblk + wn * 32 + l16;
  const int n1 = n0 + 16;
  const float bb0 = bias[n0];
  const float bb1 = bias[n1];
#pragma unroll
  for (int v = 0; v < 8; ++v) {
    const int m0 = mblk + wm * 32 + v + 8 * half;
    const int m1 = m0 + 16;
    if (m0 < M) {
      float y0 = acc00[v] + bb0;
      float y1 = acc01[v] + bb1;
      if (RELU) { y0 = fmaxf(y0, 0.f); y1 = fmaxf(y1, 0.f); }
      Y[(size_t)m0 * D_DIM + n0] = y0;
      Y[(size_t)m0 * D_DIM + n1] = y1;
    }
    if (m1 < M) {
      float y0 = acc10[v] + bb0;
      float y1 = acc11[v] + bb1;
      if (RELU) { y0 = fmaxf(y0, 0.f); y1 = fmaxf(y1, 0.f); }
      Y[(size_t)m1 * D_DIM + n0] = y0;
      Y[(size_t)m1 * D_DIM + n1] = y1;
    }
  }
}

// ---------------------------------------------------------------------------
// segment-sum: 4 edges per block, 64 lanes x float4 per edge row,
// fp32 global atomic adds (global_atomic_add_f32)
// ---------------------------------------------------------------------------
__global__ __launch_bounds__(256) void scatter_add_rows(
    const float* __restrict__ msg, const int* __restrict__ src,
    const int* __restrict__ dst, float* __restrict__ aggr, int E) {
  const int e = blockIdx.x * 4 + (threadIdx.x >> 6);
  if (e >= E) return;
  const int j = (threadIdx.x & 63) << 2;
  const int s = src[e];
  const int d = dst[e];
  const float4 v = *(const float4*)(msg + (size_t)s * D_DIM + j);
  float* base = aggr + (size_t)d * D_DIM + j;
  unsafeAtomicAdd(base + 0, v.x);
  unsafeAtomicAdd(base + 1, v.y);
  unsafeAtomicAdd(base + 2, v.z);
  unsafeAtomicAdd(base + 3, v.w);
}

__global__ __launch_bounds__(256) void zero_f32v4(float4* __restrict__ p,
                                                  long long n4) {
  long long i = (long long)blockIdx.x * 256 + threadIdx.x;
  if (i < n4) p[i] = make_float4(0.f, 0.f, 0.f, 0.f);
}

// copy a [rows,256] matrix into a strided destination (concat builder / copy)
// 4 rows per block, 64 lanes x float4 per row
__global__ __launch_bounds__(256) void copy_row_strided(
    const float* __restrict__ src, float* __restrict__ dst, int dstStride,
    int dstOff, int rows) {
  const int r = blockIdx.x * 4 + (threadIdx.x >> 6);
  const int j = (threadIdx.x & 63) << 2;
  if (r < rows) {
    *(float4*)(dst + (size_t)r * dstStride + dstOff + j) =
        *(const float4*)(src + (size_t)r * D_DIM + j);
  }
}

// ---------------------------------------------------------------------------
// host orchestration
// ---------------------------------------------------------------------------
extern "C" void kernel_launch(void* const* d_in, const int* in_sizes, int n_in,
                              void* d_out, int out_size, void* d_ws,
                              size_t ws_size, hipStream_t stream) {
  (void)n_in; (void)out_size; (void)ws_size;

  const int* l_edge = (const int*)d_in[2];
  const int* c_edge = (const int*)d_in[3];
  const float* l_emb0 = (const float*)d_in[4];
  const float* c_emb0 = (const float*)d_in[5];

  const int E = in_sizes[2];
  const int L = in_sizes[4] / D_DIM;
  const int C = in_sizes[5] / D_DIM;

  // MLP params: [0]=l2c [1]=c2l [2]=l2l [3]=cu [4]=lu ; each W1,b1,W2,b2
  const float* P[5][4];
  for (int i = 0; i < 5; ++i)
    for (int j = 0; j < 4; ++j) P[i][j] = (const float*)d_in[6 + i * 4 + j];

  float* outL = (float*)d_out;                 // [5, L, 256]
  float* outC = outL + (size_t)5 * L * D_DIM;  // [5, C, 256]

  float* ws = (float*)d_ws;
  float* hidden = ws; ws += (size_t)L * D_DIM;  // MLP hidden (max rows = L)
  float* bufMsg = ws; ws += (size_t)L * D_DIM;  // l_msg_feat / c_msg_feat
  float* bufL2L = ws; ws += (size_t)L * D_DIM;  // l2l_msg
  float* aggrC  = ws; ws += (size_t)C * D_DIM;
  float* aggrL  = ws; ws += (size_t)L * D_DIM;
  float* catC   = ws; ws += (size_t)C * 2 * D_DIM;  // [C,512]
  float* catL   = ws; ws += (size_t)L * 3 * D_DIM;  // [L,768]

  const dim3 blk(256);
  auto gemm = [&](const float* Ain, const float* W1, const float* b, float* Y,
                  int M, int K, bool relu, bool swap) {
    dim3 g((M + 127) / 128, D_DIM / 64);
    if (relu) {
      if (swap) gemm_mlp<true, true><<<g, blk, 0, stream>>>(Ain, W1, b, Y, M, K);
      else      gemm_mlp<true, false><<<g, blk, 0, stream>>>(Ain, W1, b, Y, M, K);
    } else {
      if (swap) gemm_mlp<false, true><<<g, blk, 0, stream>>>(Ain, W1, b, Y, M, K);
      else      gemm_mlp<false, false><<<g, blk, 0, stream>>>(Ain, W1, b, Y, M, K);
    }
  };
  auto rowcopy = [&](const float* src, float* dst, int stride, int off, int rows) {
    copy_row_strided<<<(rows + 3) / 4, blk, 0, stream>>>(src, dst, stride, off, rows);
  };

  // embeddings live in the output stack; slice 0 = initial values
  rowcopy(l_emb0, outL, D_DIM, 0, L);
  rowcopy(c_emb0, outC, D_DIM, 0, C);

  const long long nAggC4 = (long long)C * D_DIM / 4;
  const long long nAggL4 = (long long)L * D_DIM / 4;
  const int scatGrid = (E + 3) / 4;

  for (int it = 0; it < 4; ++it) {
    float* l_emb  = outL + (size_t)it * L * D_DIM;
    float* c_emb  = outC + (size_t)it * C * D_DIM;
    float* l_next = outL + (size_t)(it + 1) * L * D_DIM;
    float* c_next = outC + (size_t)(it + 1) * C * D_DIM;

    // l_msg_feat = MLP_l2c(l_emb)
    gemm(l_emb, P[0][0], P[0][1], hidden, L, D_DIM, true, false);
    gemm(hidden, P[0][2], P[0][3], bufMsg, L, D_DIM, false, false);

    // clause aggregation: segment_sum(l_msg_feat[l_edge], c_edge)
    zero_f32v4<<<(int)((nAggC4 + 255) / 256), blk, 0, stream>>>((float4*)aggrC, nAggC4);
    scatter_add_rows<<<scatGrid, blk, 0, stream>>>(bufMsg, l_edge, c_edge, aggrC, E);

    // clause update: c_next = MLP_cu([c_emb, aggrC])
    rowcopy(c_emb, catC, 2 * D_DIM, 0, C);
    rowcopy(aggrC, catC, 2 * D_DIM, D_DIM, C);
    gemm(catC, P[3][0], P[3][1], hidden, C, 2 * D_DIM, true, false);
    gemm(hidden, P[3][2], P[3][3], c_next, C, D_DIM, false, false);

    // c_msg_feat = MLP_c2l(OLD c_emb)
    gemm(c_emb, P[1][0], P[1][1], hidden, C, D_DIM, true, false);
    gemm(hidden, P[1][2], P[1][3], bufMsg, C, D_DIM, false, false);

    // literal aggregation: segment_sum(c_msg_feat[c_edge], l_edge)
    zero_f32v4<<<(int)((nAggL4 + 255) / 256), blk, 0, stream>>>((float4*)aggrL, nAggL4);
    scatter_add_rows<<<scatGrid, blk, 0, stream>>>(bufMsg, c_edge, l_edge, aggrL, E);

    // l2l_msg = MLP_l2l(pair-swapped l_emb)  (swap fused into A-row read)
    gemm(l_emb, P[2][0], P[2][1], hidden, L, D_DIM, true, true);
    gemm(hidden, P[2][2], P[2][3], bufL2L, L, D_DIM, false, false);

    // literal update: l_next = MLP_lu([l_emb, aggrL, l2l_msg])
    rowcopy(l_emb, catL, 3 * D_DIM, 0, L);
    rowcopy(aggrL, catL, 3 * D_DIM, D_DIM, L);
    rowcopy(bufL2L, catL, 3 * D_DIM, 2 * D_DIM, L);
    gemm(catL, P[4][0], P[4][1], hidden, L, 3 * D_DIM, true, false);
    gemm(hidden, P[4][2], P[4][3], l_next, L, D_DIM, false, false);
  }
}